// GCNLayer_89730456748328
// MI455X (gfx1250) — compile-verified
//
#include <hip/hip_runtime.h>
#include <hip/hip_bf16.h>

// ---------------------------------------------------------------------------
// GCN layer for MI455X (gfx1250, wave32).
//  Stage 1: edge scatter (segment sum / max / count) with L2-resident atomics.
//  Stage 2: fused per-node MLP using V_WMMA_F32_16X16X4_F32 (f32 matrix core).
//  Weights are pre-transposed AND pair-interleaved so every WMMA B-fragment
//  is a single global_load_b64.
// ---------------------------------------------------------------------------

typedef __attribute__((ext_vector_type(2))) float v2f;
typedef __attribute__((ext_vector_type(8))) float v8f;

#define FDIM 128
#define BDIM 2
#define NEG_INF_BITS 0xFF800000u
#define LDSROW 130   // 128 + 2 pad: A-fragment rows land on distinct LDS banks

// ---------------- init: sum=0, max=-inf, cnt=0 -----------------------------
__global__ void gcn_init_kernel(float* __restrict__ sumb,
                                unsigned* __restrict__ maxb,
                                float* __restrict__ cnt,
                                long long MF, int N) {
    long long i = (long long)blockIdx.x * blockDim.x + threadIdx.x;
    if (i < MF) {
        sumb[i] = 0.0f;
        maxb[i] = NEG_INF_BITS;
    }
    if (i < N) cnt[i] = 0.0f;
}

// Sign-aware float atomic max via integer atomics (valid for all finite
// values given the -inf initialization; int-max handles >=0, uint-min <0).
__device__ __forceinline__ void atomicMaxF(float* addr, float v) {
    if (v >= 0.0f) atomicMax((int*)addr, __float_as_int(v));
    else           atomicMin((unsigned int*)addr, __float_as_uint(v));
}

// ---------------- scatter: one edge per 32-lane group ----------------------
// lane handles 4 consecutive features (float4) for both batch entries.
__global__ void gcn_scatter_kernel(const float* __restrict__ x0,
                                   const int* __restrict__ dst,
                                   const int* __restrict__ src,
                                   float* __restrict__ sumb,
                                   float* __restrict__ maxb,
                                   float* __restrict__ cnt,
                                   int E, int N) {
    long long tid = (long long)blockIdx.x * blockDim.x + threadIdx.x;
    int e    = (int)(tid >> 5);
    int lane = (int)(tid & 31);
    if (e >= E) return;
    int s = src[e];
    int d = dst[e];
    int f = lane * 4;
#pragma unroll
    for (int b = 0; b < BDIM; ++b) {
        const float4 v = *(const float4*)(x0 + ((long long)b * N + s) * FDIM + f);
        float* sp = sumb + ((long long)d * BDIM + b) * FDIM + f;
        float* mp = maxb + ((long long)d * BDIM + b) * FDIM + f;
        atomicAdd(sp + 0, v.x); atomicAdd(sp + 1, v.y);
        atomicAdd(sp + 2, v.z); atomicAdd(sp + 3, v.w);
        atomicMaxF(mp + 0, v.x); atomicMaxF(mp + 1, v.y);
        atomicMaxF(mp + 2, v.z); atomicMaxF(mp + 3, v.w);
    }
    if (lane == 0) atomicAdd(cnt + d, 1.0f);
}

// ---------------- prep: transpose + pair-interleave weights ----------------
// w0p[p][o][j] = w0[o][2p+j]   p<128, o<128, j<2   (fragment-ready layout)
// w1p[p][o][j] = w1[o][2p+j]   p<64
// A WMMA B-fragment (bf.x = W[k0][col], bf.y = W[k0+1][col]) is then the
// 8-byte word at w*p + (p*256 + col*2), i.e. one global_load_b64.
__global__ void gcn_prep_kernel(const float* __restrict__ w0,
                                const float* __restrict__ w1,
                                float* __restrict__ w0p,
                                float* __restrict__ w1p) {
    int i = blockIdx.x * blockDim.x + threadIdx.x;
    const int W0N = 2 * FDIM * FDIM;  // 32768
    const int W1N = FDIM * FDIM;      // 16384
    if (i < W0N) {
        int k = i / FDIM, o = i % FDIM;
        w0p[(k >> 1) * (2 * FDIM) + o * 2 + (k & 1)] = w0[o * (2 * FDIM) + k];
    } else if (i < W0N + W1N) {
        int j = i - W0N;
        int k = j / FDIM, o = j % FDIM;
        w1p[(k >> 1) * (2 * FDIM) + o * 2 + (k & 1)] = w1[o * FDIM + k];
    }
}

// ---------------- fused MLP: one wave computes a 16-row x 128-col tile -----
__global__ __launch_bounds__(128)
void gcn_mlp_kernel(const float* __restrict__ x0,
                    const float* __restrict__ sumb,
                    const float* __restrict__ maxb,
                    const float* __restrict__ cnt,
                    const float* __restrict__ w0p,
                    const float* __restrict__ w1p,
                    const float* __restrict__ b0,
                    const float* __restrict__ b1,
                    float* __restrict__ out,
                    int N, int M, int ntiles) {
    __shared__ float hbuf[4][16 * LDSROW];   // per-wave h staging (~32.5 KB)

    const int wv   = threadIdx.x >> 5;
    const int lane = threadIdx.x & 31;
    const int tile = blockIdx.x * 4 + wv;
    if (tile >= ntiles) return;              // wave-uniform: EXEC stays all-1s

    const int rowBase = tile * 16;
    const int lr   = lane & 15;              // column / M-row selector
    const int half = lane >> 4;              // K sub-step selector (ISA layout)

    // A-fragment source row for this lane (lanes 16-31 mirror lanes 0-15)
    int row = rowBase + lr;
    if (row >= M) row = M - 1;
    const int nn = row >> 1;                 // row = n*B + b, B==2
    const int bb = row & 1;

    // Fused mean / isolated-node fallback (reference: where(has_edge, s/cnt, x0))
    const float c     = cnt[nn];
    const bool  hasE  = c > 0.0f;
    const float scale = hasE ? (1.0f / c) : 1.0f;
    const float* x0row = x0 + ((long long)bb * N + nn) * FDIM;
    const float* meanp = hasE ? (sumb + (long long)row * FDIM) : x0row;
    const float* maxp  = hasE ? (maxb + (long long)row * FDIM) : x0row;

    // per-lane fragment base pointers (k0 = s*4 + 2*half, pair p = 2s + half)
    const float* mean_l = meanp + half * 2;          // + s*4 per step
    const float* max_l  = maxp  + half * 2;
    const float* w0_l   = w0p + (long long)half * (2 * FDIM) + lr * 2; // + s*512
    const float* w1_l   = w1p + (long long)half * (2 * FDIM) + lr * 2;

    // ---------------- GEMM1: h = relu([mean||max] @ w0^T + b0), K = 256 ----
    v8f acc[8] = {};
    for (int s = 0; s < 32; ++s) {           // K 0..127: mean half (branch-free)
        v2f a = *(const v2f*)(mean_l + s * 4);
        a.x *= scale; a.y *= scale;
        const float* wk = w0_l + s * (4 * FDIM);
#pragma unroll
        for (int ct = 0; ct < 8; ++ct) {
            v2f bf = *(const v2f*)(wk + ct * 32);
            acc[ct] = __builtin_amdgcn_wmma_f32_16x16x4_f32(
                false, a, false, bf, (short)0, acc[ct], false, false);
        }
    }
    for (int s = 0; s < 32; ++s) {           // K 128..255: max half (branch-free)
        v2f a = *(const v2f*)(max_l + s * 4);
        const float* wk = w0_l + (32 + s) * (4 * FDIM);
#pragma unroll
        for (int ct = 0; ct < 8; ++ct) {
            v2f bf = *(const v2f*)(wk + ct * 32);
            acc[ct] = __builtin_amdgcn_wmma_f32_16x16x4_f32(
                false, a, false, bf, (short)0, acc[ct], false, false);
        }
    }

    // bias + relu, then spill h to LDS (C-layout -> row-major, padded rows)
    float* hb = hbuf[wv];
#pragma unroll
    for (int ct = 0; ct < 8; ++ct) {
        const float bv = b0[ct * 16 + lr];
#pragma unroll
        for (int r = 0; r < 8; ++r) {
            float t = acc[ct][r] + bv;
            t = t > 0.0f ? t : 0.0f;
            hb[(r + 8 * half) * LDSROW + ct * 16 + lr] = t;
        }
    }

    // ---------------- GEMM2: out = x0 + h @ w1^T + b1, K = 128 -------------
    const float* hb_l = hb + lr * LDSROW + half * 2;
    v8f acc2[8] = {};
    for (int s = 0; s < 32; ++s) {
        v2f a = *(const v2f*)(hb_l + s * 4);             // 8B-aligned ds read
        const float* wk = w1_l + s * (4 * FDIM);
#pragma unroll
        for (int ct = 0; ct < 8; ++ct) {
            v2f bf = *(const v2f*)(wk + ct * 32);
            acc2[ct] = __builtin_amdgcn_wmma_f32_16x16x4_f32(
                false, a, false, bf, (short)0, acc2[ct], false, false);
        }
    }

    // epilogue: residual + bias, store [B, N, F]
#pragma unroll
    for (int ct = 0; ct < 8; ++ct) {
        const float bv = b1[ct * 16 + lr];
#pragma unroll
        for (int r = 0; r < 8; ++r) {
            const int orow = rowBase + r + 8 * half;
            if (orow < M) {
                const int on = orow >> 1, ob = orow & 1;
                const long long idx = ((long long)ob * N + on) * FDIM + ct * 16 + lr;
                out[idx] = x0[idx] + bv + acc2[ct][r];
            }
        }
    }
}

// ---------------------------------------------------------------------------
extern "C" void kernel_launch(void* const* d_in, const int* in_sizes, int n_in,
                              void* d_out, int out_size, void* d_ws, size_t ws_size,
                              hipStream_t stream) {
    const float* x0  = (const float*)d_in[0];
    const int*   dst = (const int*)d_in[1];
    const int*   src = (const int*)d_in[2];
    const float* w0  = (const float*)d_in[3];
    const float* b0  = (const float*)d_in[4];
    const float* w1  = (const float*)d_in[5];
    const float* b1  = (const float*)d_in[6];
    float* out = (float*)d_out;

    const int E = in_sizes[1];
    const int N = in_sizes[0] / (BDIM * FDIM);
    const long long M  = (long long)N * BDIM;
    const long long MF = M * FDIM;

    // workspace layout: sum | max | cnt | w0p | w1p   (~103 MB)
    float* sumb = (float*)d_ws;
    float* maxb = sumb + MF;
    float* cnt  = maxb + MF;
    float* w0p  = cnt + N;
    float* w1p  = w0p + 2 * FDIM * FDIM;

    {   // init
        long long total = MF;
        int blocks = (int)((total + 255) / 256);
        gcn_init_kernel<<<blocks, 256, 0, stream>>>(sumb, (unsigned*)maxb, cnt, MF, N);
    }
    {   // scatter (atomics, L2-resident)
        long long threads = (long long)E * 32;
        int blocks = (int)((threads + 255) / 256);
        gcn_scatter_kernel<<<blocks, 256, 0, stream>>>(x0, dst, src, sumb, maxb, cnt, E, N);
    }
    {   // weight transpose + pair interleave
        int total = 2 * FDIM * FDIM + FDIM * FDIM;
        gcn_prep_kernel<<<(total + 255) / 256, 256, 0, stream>>>(w0, w1, w0p, w1p);
    }
    {   // fused WMMA MLP
        int ntiles = (int)((M + 15) / 16);
        int blocks = (ntiles + 3) / 4;
        gcn_mlp_kernel<<<blocks, 128, 0, stream>>>(x0, sumb, maxb, cnt, w0p, w1p,
                                                   b0, b1, out, N, (int)M, ntiles);
    }
}